// FCHLCuda_23124103922413
// MI455X (gfx1250) — compile-verified
//
#include <hip/hip_runtime.h>
#include <hip/hip_bf16.h>

#define NSPEC   4
#define NRS2    24
#define NRS3    20
#define NPAIR   10
#define NATMAX  28
#define FEAT    496           // 4*24 + 10*20*2
#define NPAIRS2 (NATMAX*NATMAX)        // 784
#define NTRIP   378                    // C(28,2)
#define RCUTF   8.0f
#define ETA2F   0.32f
#define ETA3F   2.7f

typedef __attribute__((ext_vector_type(2))) float v2f;
typedef __attribute__((ext_vector_type(8))) float v8f;

__global__ __launch_bounds__(512) void fchl19_kernel(
    const float* __restrict__ X,          // [B][28][3]
    const float* __restrict__ Z,          // [B][28]
    const int*   __restrict__ atom_counts,// [B]
    float*       __restrict__ out)        // [B][28][496]
{
  __shared__ float s_out[NATMAX * FEAT];  // 55552 B; first 1024 floats reused as Gram scratch
  __shared__ float s_r [NATMAX][NATMAX];
  __shared__ float s_fc[NATMAX][NATMAX];
  __shared__ float s_x[32], s_y[32], s_z[32];
  __shared__ int   s_spec[32];
  __shared__ unsigned char s_pj[NTRIP], s_pk[NTRIP];
  __shared__ float s_rs2inv[NRS2], s_lrs2[NRS2];

  const int b    = blockIdx.x;
  const int tid  = threadIdx.x;
  const int nthr = blockDim.x;
  const int natoms = atom_counts[b];
  float* sG = s_out;                      // 32x32 Gram scratch (aliased, pre-zeroing)

  // ---- Phase 0: stage atom data into LDS --------------------------------
  // Coordinates are centered by -3.0 (translation-invariant) to reduce
  // |x|^2 and hence cancellation error in the Gram-based distance.
  if (tid < 32) {
    float x = 0.f, y = 0.f, z = 0.f; int sp = -1;
    if (tid < NATMAX) {
      x = X[(b*NATMAX + tid)*3 + 0] - 3.0f;
      y = X[(b*NATMAX + tid)*3 + 1] - 3.0f;
      z = X[(b*NATMAX + tid)*3 + 2] - 3.0f;
      float zv = Z[b*NATMAX + tid];
      sp = (zv == 1.0f) ? 0 : (zv == 6.0f) ? 1 : (zv == 7.0f) ? 2 : (zv == 8.0f) ? 3 : -1;
    }
    s_x[tid] = x; s_y[tid] = y; s_z[tid] = z; s_spec[tid] = sp;
  }
  if (tid >= 32 && tid < 32 + NRS2) {
    const int m = tid - 32;
    float rs2 = (float)((m + 1) * (8.0 / 24.0));
    s_rs2inv[m] = 1.0f / rs2;
    s_lrs2[m]   = logf(rs2);
  }
  __syncthreads();

  // ---- Phase 1: Gram matrix X*X^T via V_WMMA_F32_16X16X4_F32 (wave 0) ---
  if (tid < 32) {                         // full wave 0: EXEC all ones
    const int lane = tid;
    const int half = lane >> 4;           // 0: K=0,1   1: K=2,3
    const int lm   = lane & 15;
    #pragma unroll
    for (int ti = 0; ti < 2; ++ti) {
      v2f a;
      const int ra = ti*16 + lm;
      a[0] = half ? s_z[ra] : s_x[ra];
      a[1] = half ? 0.0f    : s_y[ra];
      #pragma unroll
      for (int tj = 0; tj < 2; ++tj) {
        v2f bv;
        const int cb = tj*16 + lm;
        bv[0] = half ? s_z[cb] : s_x[cb];
        bv[1] = half ? 0.0f    : s_y[cb];
        v8f acc = {};
        acc = __builtin_amdgcn_wmma_f32_16x16x4_f32(
                /*neg_a=*/false, a, /*neg_b=*/false, bv,
                /*c_mod=*/(short)0, acc, /*reuse_a=*/false, /*reuse_b=*/false);
        #pragma unroll
        for (int rr = 0; rr < 8; ++rr) {
          const int row = ti*16 + rr + 8*half;
          const int col = tj*16 + lm;
          if (row < NATMAX && col < NATMAX) sG[row*32 + col] = acc[rr];
        }
      }
    }
  }
  __syncthreads();

  // ---- Phase 2: distances, cutoff function, triplet LUT ------------------
  for (int p = tid; p < NPAIRS2; p += nthr) {
    const int i = p / NATMAX, j = p % NATMAX;
    float d2 = sG[i*32 + i] + sG[j*32 + j] - 2.0f * sG[i*32 + j];
    d2 = fmaxf(d2, 1e-12f);
    const float rv = sqrtf(d2);
    const bool within = (i != j) && (i < natoms) && (j < natoms) && (rv < RCUTF);
    s_r [i][j] = rv;
    s_fc[i][j] = within ? 0.5f * (cosf(rv * (3.14159265358979323846f / 8.0f)) + 1.0f) : 0.0f;
  }
  if (tid < NATMAX) {
    const int j = tid;
    const int base = 27*j - (j*(j-1))/2;
    for (int k = j + 1; k < NATMAX; ++k) {
      const int idx = base + (k - j - 1);
      s_pj[idx] = (unsigned char)j;
      s_pk[idx] = (unsigned char)k;
    }
  }
  __syncthreads();

  // ---- Phase 3: zero the accumulator (Gram scratch no longer needed) ----
  {
    float4* s4 = (float4*)s_out;
    const float4 z4 = {0.f, 0.f, 0.f, 0.f};
    for (int p = tid; p < (NATMAX*FEAT)/4; p += nthr) s4[p] = z4;
  }
  __syncthreads();

  // ---- Phase 4a: two-body terms -----------------------------------------
  const float INV_SQRT_2PI = 0.3989422804014327f;
  for (int p = tid; p < NPAIRS2; p += nthr) {
    const int i = p / NATMAX, j = p % NATMAX;
    const float fcij = s_fc[i][j];
    if (fcij <= 0.0f) continue;
    const float rv     = s_r[i][j];
    const float lr     = logf(rv);
    const float ln     = log1pf(ETA2F / (rv*rv));
    const float mu     = lr - 0.5f * ln;
    const float inv2ln = 1.0f / (2.0f * ln);
    const float pref   = fcij * expf(-1.8f * lr) * INV_SQRT_2PI * (1.0f / sqrtf(ln));
    const int   sj     = s_spec[j];
    float* dst = &s_out[i*FEAT + sj*NRS2];
    #pragma unroll 4
    for (int m = 0; m < NRS2; ++m) {
      const float d = s_lrs2[m] - mu;
      atomicAdd(&dst[m], pref * s_rs2inv[m] * expf(-d*d*inv2ln));
    }
  }

  // ---- Phase 4b: three-body terms ---------------------------------------
  const float W3 = 12.4225786f;  // sqrt(ETA3/pi)*13.4
  for (int wk = tid; wk < NATMAX*NTRIP; wk += nthr) {
    const int i = wk / NTRIP;
    const int p = wk % NTRIP;
    const int j = s_pj[p], k = s_pk[p];
    const float fij = s_fc[i][j];
    const float fik = s_fc[i][k];
    if (fij <= 0.0f || fik <= 0.0f) continue;
    const float rij = s_r[i][j], rik = s_r[i][k], rjk = s_r[j][k];
    const float rij2 = rij*rij, rik2 = rik*rik, rjk2 = rjk*rjk;
    const float cos_i = (rij2 + rik2 - rjk2) / (2.0f * rij * rik);
    const float cos_j = (rij2 + rjk2 - rik2) / (2.0f * rij * rjk);
    const float cos_k = (rik2 + rjk2 - rij2) / (2.0f * rik * rjk);
    const float atm   = (1.0f + 3.0f*cos_i*cos_j*cos_k)
                      * expf(-0.57f * logf(rij*rik*rjk));
    const float wgt   = W3 * atm * fij * fik;
    const float cc = fminf(fmaxf(cos_i, -1.0f), 1.0f);
    const float ss = sqrtf(fminf(fmaxf(1.0f - cc*cc, 1e-12f), 1.0f));
    const float a0 = 2.0f * cc * wgt;
    const float a1 = 2.0f * ss * wgt;
    const int sj = s_spec[j], sk = s_spec[k];
    const int pmin = sj < sk ? sj : sk;
    const int pmax = sj < sk ? sk : sj;
    const int P = pmin*4 - (pmin*(pmin+1))/2 + pmax;   // upper-tri pair class 0..9
    const float mid = 0.5f * (rij + rik);
    float* dst = &s_out[i*FEAT + NSPEC*NRS2 + P*(NRS3*2)];
    #pragma unroll 4
    for (int l = 0; l < NRS3; ++l) {
      const float dr = mid - (float)((l + 1) * (8.0 / 20.0));
      const float e  = expf(-ETA3F * dr * dr);
      atomicAdd(&dst[2*l + 0], a0 * e);
      atomicAdd(&dst[2*l + 1], a1 * e);
    }
  }
  __syncthreads();

  // ---- Phase 5: writeout (vectorized, coalesced) -------------------------
  {
    const float4* s4 = (const float4*)s_out;
    float4* dst = (float4*)(out + (size_t)b * NATMAX * FEAT);
    for (int p = tid; p < (NATMAX*FEAT)/4; p += nthr) dst[p] = s4[p];
  }
}

extern "C" void kernel_launch(void* const* d_in, const int* in_sizes, int n_in,
                              void* d_out, int out_size, void* d_ws, size_t ws_size,
                              hipStream_t stream) {
  (void)n_in; (void)out_size; (void)d_ws; (void)ws_size;
  const float* X  = (const float*)d_in[0];
  const float* Z  = (const float*)d_in[1];
  const int* atom_counts = (const int*)d_in[4];
  const int B = in_sizes[4];              // 64 molecules
  fchl19_kernel<<<B, 512, 0, stream>>>(X, Z, atom_counts, (float*)d_out);
}